// YOLOv4Loss_11527692223112
// MI455X (gfx1250) — compile-verified
//
#include <hip/hip_runtime.h>
#include <math.h>

#define NCLS 5
#define EPSF 1e-7f
#define FOUR_OVER_PI2 0.40528473456935108577f

typedef __attribute__((ext_vector_type(2))) float v2f;
typedef __attribute__((ext_vector_type(8))) float v8f;

__device__ __forceinline__ float softplusf(float x) {
    // stable: max(x,0) + log1p(exp(-|x|))
    return fmaxf(x, 0.0f) + log1pf(expf(-fabsf(x)));
}

__device__ __forceinline__ float sigmoidf(float x) {
    return 1.0f / (1.0f + expf(-x));
}

// Sum a value over all 32 lanes of the wave using V_WMMA_F32_16X16X4_F32.
// A = ones(16x4), B packs (v, 0) per lane -> D[m][n] = colsum(n), and
// sum_n D[0][n] over a 16-lane half = sum of all 64 B slots = wave total.
// Layout-agnostic: every B slot contributes exactly once to the grand total.
// Requires EXEC == all ones (callers have no divergence at this point).
__device__ __forceinline__ float wave_sum32_wmma(float v) {
    v2f a; a[0] = 1.0f; a[1] = 1.0f;
    v2f b; b[0] = v;    b[1] = 0.0f;
    v8f c = {};
    v8f d = __builtin_amdgcn_wmma_f32_16x16x4_f32(false, a, false, b,
                                                  (short)0, c, false, false);
    float s = d[0];                 // lanes 0-15: D[0][n]; lanes 16-31: D[8][n] (rows identical)
    s += __shfl_xor(s, 1, 32);
    s += __shfl_xor(s, 2, 32);
    s += __shfl_xor(s, 4, 32);
    s += __shfl_xor(s, 8, 32);      // xor masks 1..8 stay within each 16-lane half
    return s;                       // wave total, broadcast to all lanes
}

// One block: 256 threads, covers a chunk of H*W cells for batch b = blockIdx.y.
// partials[(blockIdx.y*gridDim.x + blockIdx.x)*5 + q] = block sums of
// q in {coord, obj, noobj, cls, npos}.
__global__ void __launch_bounds__(256)
yolo_scale_kernel(const float* __restrict__ pred,      // [B, 30, H, W]
                  const float* __restrict__ targets,   // [B, N, 5]
                  const float* __restrict__ anc,       // &anchors[s][0][0] (2 floats)
                  int B, int N, int H, int W,
                  float* __restrict__ partials)
{
    const int b   = blockIdx.y;
    const int HW  = H * W;
    const int tid = threadIdx.x;

    __shared__ int   s_cell[32];
    __shared__ float s_tx[32], s_ty[32], s_tw[32], s_th[32];
    __shared__ int   s_cls[32];
    __shared__ float s_red[8][5];

    if (tid < N) {
        const float* t = targets + ((size_t)b * N + tid) * 5;
        float bx = t[0], by = t[1];
        int gx = (int)floorf(bx * (float)W);
        int gy = (int)floorf(by * (float)H);
        bool valid = (gx >= 0) && (gx < W) && (gy >= 0) && (gy < H);
        s_cell[tid] = valid ? (gy * W + gx) : -1;
        s_tx[tid] = t[0]; s_ty[tid] = t[1]; s_tw[tid] = t[2]; s_th[tid] = t[3];
        s_cls[tid] = (int)t[4];
    }
    __syncthreads();

    const float ancx = anc[0];
    const float ancy = anc[1];

    float acc_coord = 0.0f, acc_obj = 0.0f, acc_noobj = 0.0f;
    float acc_cls = 0.0f, acc_npos = 0.0f;

    const int hw = blockIdx.x * blockDim.x + tid;
    if (hw < HW) {
        const float* pb = pred + (size_t)b * 30 * HW;   // 3 anchors * (5+NCLS) channels

        // objectness for the 3 anchors (the only channels needed everywhere)
        float o0 = pb[(size_t)( 4) * HW + hw];
        float o1 = pb[(size_t)(14) * HW + hw];
        float o2 = pb[(size_t)(24) * HW + hw];

        // which target box (if any) maps to this cell; last valid match wins
        int best = -1;
        for (int n = 0; n < N; ++n)
            if (s_cell[n] == hw) best = n;

        acc_noobj = softplusf(o1) + softplusf(o2);
        if (best < 0) {
            acc_noobj += softplusf(o0);
        } else {
            // anchor-0 box + class channels, only at positive cells (rare)
            float p0 = pb[(size_t)0 * HW + hw];
            float p1 = pb[(size_t)1 * HW + hw];
            float p2 = pb[(size_t)2 * HW + hw];
            float p3 = pb[(size_t)3 * HW + hw];
            float c0 = pb[(size_t)5 * HW + hw];
            float c1 = pb[(size_t)6 * HW + hw];
            float c2 = pb[(size_t)7 * HW + hw];
            float c3 = pb[(size_t)8 * HW + hw];
            float c4 = pb[(size_t)9 * HW + hw];

            int   wi = hw % W, hi = hw / W;
            float px = (sigmoidf(p0) + (float)wi) / (float)W;
            float py = (sigmoidf(p1) + (float)hi) / (float)H;
            float pw = expf(p2) * ancx * (1.0f / 800.0f);
            float ph = expf(p3) * ancy * (1.0f / 800.0f);

            float tx = s_tx[best], ty = s_ty[best];
            float tw = s_tw[best], th = s_th[best];

            // CIoU
            float b1x1 = px - 0.5f * pw, b1x2 = px + 0.5f * pw;
            float b1y1 = py - 0.5f * ph, b1y2 = py + 0.5f * ph;
            float b2x1 = tx - 0.5f * tw, b2x2 = tx + 0.5f * tw;
            float b2y1 = ty - 0.5f * th, b2y2 = ty + 0.5f * th;
            float iw = fmaxf(fminf(b1x2, b2x2) - fmaxf(b1x1, b2x1), 0.0f);
            float ih = fmaxf(fminf(b1y2, b2y2) - fmaxf(b1y1, b2y1), 0.0f);
            float inter = iw * ih;
            float uni   = pw * ph + tw * th - inter;
            float iou   = inter / (uni + EPSF);
            float ex = fmaxf(b1x2, b2x2) - fminf(b1x1, b2x1);
            float ey = fmaxf(b1y2, b2y2) - fminf(b1y1, b2y1);
            float cd = (px - tx) * (px - tx) + (py - ty) * (py - ty);
            float diou = iou - cd / (ex * ex + ey * ey + EPSF);
            float dat  = atanf(tw / (th + EPSF)) - atanf(pw / (ph + EPSF));
            float vv   = FOUR_OVER_PI2 * dat * dat;
            float alpha = vv / (1.0f - iou + vv + EPSF);
            float ciou  = diou - alpha * vv;

            acc_coord = 1.0f - ciou;
            acc_obj   = softplusf(-o0);
            acc_npos  = 1.0f;

            // class NLL = logsumexp - logit[cls]
            float m = fmaxf(fmaxf(fmaxf(c0, c1), fmaxf(c2, c3)), c4);
            float lse = m + logf(expf(c0 - m) + expf(c1 - m) + expf(c2 - m) +
                                 expf(c3 - m) + expf(c4 - m));
            float sel = (s_cls[best] == 0) ? c0 :
                        (s_cls[best] == 1) ? c1 :
                        (s_cls[best] == 2) ? c2 :
                        (s_cls[best] == 3) ? c3 : c4;
            acc_cls = lse - sel;
        }
    }

    // wave-level reduction via WMMA (all 256 threads reach here; EXEC all ones)
    float r0 = wave_sum32_wmma(acc_coord);
    float r1 = wave_sum32_wmma(acc_obj);
    float r2 = wave_sum32_wmma(acc_noobj);
    float r3 = wave_sum32_wmma(acc_cls);
    float r4 = wave_sum32_wmma(acc_npos);

    const int wave = tid >> 5;
    if ((tid & 31) == 0) {
        s_red[wave][0] = r0; s_red[wave][1] = r1; s_red[wave][2] = r2;
        s_red[wave][3] = r3; s_red[wave][4] = r4;
    }
    __syncthreads();
    if (tid < 5) {
        float s = 0.0f;
        for (int w8 = 0; w8 < 8; ++w8) s += s_red[w8][tid];
        partials[((size_t)blockIdx.y * gridDim.x + blockIdx.x) * 5 + tid] = s;
    }
}

// Single block; deterministic fixed-order summation of block partials,
// then final loss combination (COORD_S=1, OBJ_S=1, NOOBJ_S=0.5, CLS_S=1).
__global__ void yolo_finalize(const float* __restrict__ ws,
                              int nb0, int nb1, int nb2,
                              int B, float* __restrict__ out)
{
    __shared__ float sums[3][5];
    const int tid = threadIdx.x;
    if (tid < 15) {
        int s = tid / 5, q = tid % 5;
        int nb[3]  = {nb0, nb1, nb2};
        int off[3] = {0, nb0 * 5, (nb0 + nb1) * 5};
        const float* p = ws + off[s];
        float acc = 0.0f;
        for (int i = 0; i < nb[s]; ++i) acc += p[i * 5 + q];
        sums[s][q] = acc;
    }
    __syncthreads();
    if (tid == 0) {
        const int Hs[3] = {100, 50, 25};
        float tc = 0.0f, to = 0.0f, tn = 0.0f, tk = 0.0f;
        for (int s = 0; s < 3; ++s) {
            float coordsum = sums[s][0], objsum = sums[s][1];
            float noobjsum = sums[s][2], clssum = sums[s][3];
            float npos = sums[s][4];
            float safe = fmaxf(npos, 1.0f);
            bool  hp   = npos > 0.0f;
            float cl = hp ? (coordsum / safe) : 0.0f;
            float ol = hp ? (objsum   / safe) : 0.0f;
            float kl = hp ? (clssum   / safe) : 0.0f;
            float cnt = (float)B * 3.0f * (float)(Hs[s] * Hs[s]) - npos;
            float nl = noobjsum / cnt;
            tc += cl; to += ol; tn += 0.5f * nl; tk += kl;
        }
        out[0] = tc + to + tn + tk;
        out[1] = tc; out[2] = to; out[3] = tn; out[4] = tk;
    }
}

extern "C" void kernel_launch(void* const* d_in, const int* in_sizes, int n_in,
                              void* d_out, int out_size, void* d_ws, size_t ws_size,
                              hipStream_t stream) {
    const float* preds[3] = {(const float*)d_in[0], (const float*)d_in[1],
                             (const float*)d_in[2]};
    const float* targets = (const float*)d_in[3];
    const float* anchors = (const float*)d_in[4];
    float* out = (float*)d_out;
    float* ws  = (float*)d_ws;

    const int Hs[3] = {100, 50, 25};
    const int B = in_sizes[0] / (30 * 100 * 100);   // 32
    const int N = in_sizes[3] / (B * 5);            // 20

    int nb[3];
    int chunks[3];
    for (int s = 0; s < 3; ++s) {
        int HW = Hs[s] * Hs[s];
        chunks[s] = (HW + 255) / 256;
        nb[s] = chunks[s] * B;                      // blocks per scale
    }
    float* part[3];
    part[0] = ws;
    part[1] = ws + (size_t)nb[0] * 5;
    part[2] = ws + (size_t)(nb[0] + nb[1]) * 5;

    for (int s = 0; s < 3; ++s) {
        dim3 grid(chunks[s], B);
        yolo_scale_kernel<<<grid, 256, 0, stream>>>(
            preds[s], targets, anchors + s * 6, B, N, Hs[s], Hs[s], part[s]);
    }
    yolo_finalize<<<1, 32, 0, stream>>>(ws, nb[0], nb[1], nb[2], B, out);
}